// FDConvClassifier_21646635172388
// MI455X (gfx1250) — compile-verified
//
#include <hip/hip_runtime.h>
#include <math.h>

// FDConv frequency-band attention for MI455X (gfx1250, wave32).
// low_k = D_k * X * D_k  (D_k = 64x64 symmetric circulant Dirichlet matrix),
// out = s1*x + (s2-s1)*low1 + (s3-s2)*low2 + (1-s3)*low3,  s_i = 2*sigmoid(conv_i(x)).
// f32 precision preserved end-to-end via V_WMMA_F32_16X16X4_F32.
// LDS matrices padded to 68-float row stride: conflict-free, 8B-aligned b64 DS ops.

typedef __attribute__((ext_vector_type(2))) float v2f;
typedef __attribute__((ext_vector_type(8))) float v8f;
typedef __attribute__((ext_vector_type(4))) float v4f;

#define HW 64
#define PIX 4096           // 64*64
#define BATCH 32
#define CHAN 256
#define GROUPS 8
#define CPG 32             // channels per group
#define NMAP (BATCH*GROUPS*PIX)   // one coefficient plane: 1,048,576 floats
#define SX 68              // padded LDS row stride (bank-conflict-free, 8B-aligned)

// ---------------------------------------------------------------- K0: D matrices
__global__ void build_dmat_kernel(float* __restrict__ dD) {
    int idx = blockIdx.x * blockDim.x + threadIdx.x;   // 3*4096 threads
    if (idx >= 3 * PIX) return;
    int band = idx / PIX;
    int rem  = idx % PIX;
    int i = rem / HW, j = rem % HW;
    int M = (band == 0) ? 16 : (band == 1) ? 8 : 4;    // k = 2,4,8 -> cutoff bins
    float d = (float)(i - j);
    float acc = 1.0f;
    for (int m = 1; m <= M; ++m)
        acc += 2.0f * __cosf(6.2831853071795864f * (float)m * d * (1.0f / 64.0f));
    dD[idx] = acc * (1.0f / 64.0f);
}

// ---------------------------------------------------------------- K1: grouped conv -> coeff maps
// grid: (b*8+g)*16 + tile   (16 tiles of 16x16 pixels), 256 threads
__global__ void conv_coeff_kernel(const float* __restrict__ x,
                                  const float* __restrict__ w,     // [3][8][32][3][3]
                                  const float* __restrict__ bias,  // [3][8]
                                  float* __restrict__ cmap) {      // [4][32*8][4096]
    __shared__ float wlds[3 * CPG * 9];    // 864 floats
    __shared__ float tile[18 * 18];        // halo tile for one channel

    const int tid  = threadIdx.x;
    const int bg   = blockIdx.x >> 4;
    const int t    = blockIdx.x & 15;
    const int b    = bg >> 3;
    const int g    = bg & 7;
    const int oy   = (t >> 2) * 16;
    const int ox   = (t & 3) * 16;
    const int ty   = tid >> 4;
    const int tx   = tid & 15;

    // stage this group's weights for all 3 bands
    for (int i = tid; i < 3 * CPG * 9; i += 256) {
        int band = i / (CPG * 9);
        int rem  = i % (CPG * 9);
        wlds[i] = w[((band * GROUPS + g) * CPG) * 9 + rem];
    }

    float f0 = bias[0 * GROUPS + g];
    float f1 = bias[1 * GROUPS + g];
    float f2 = bias[2 * GROUPS + g];

    const float* xg = x + (size_t)(b * CHAN + g * CPG) * PIX;

    for (int ci = 0; ci < CPG; ++ci) {
        __syncthreads();                       // previous tile fully consumed
        const float* xc = xg + (size_t)ci * PIX;
        for (int i = tid; i < 18 * 18; i += 256) {
            int r = i / 18, c = i % 18;
            int yy = oy + r - 1, xx = ox + c - 1;
            tile[i] = (yy >= 0 && yy < HW && xx >= 0 && xx < HW) ? xc[yy * HW + xx] : 0.0f;
        }
        __syncthreads();
        const float* wr = &wlds[ci * 9];
        #pragma unroll
        for (int dy = 0; dy < 3; ++dy) {
            #pragma unroll
            for (int dx = 0; dx < 3; ++dx) {
                float v = tile[(ty + dy) * 18 + (tx + dx)];
                f0 = fmaf(wr[dy * 3 + dx],                 v, f0);
                f1 = fmaf(wr[CPG * 9 + dy * 3 + dx],       v, f1);
                f2 = fmaf(wr[2 * CPG * 9 + dy * 3 + dx],   v, f2);
            }
        }
    }

    float s1 = 2.0f / (1.0f + __expf(-f0));
    float s2 = 2.0f / (1.0f + __expf(-f1));
    float s3 = 2.0f / (1.0f + __expf(-f2));

    int base = (b * GROUPS + g) * PIX + (oy + ty) * HW + (ox + tx);
    cmap[0 * NMAP + base] = s1;         // multiplies x
    cmap[1 * NMAP + base] = s2 - s1;    // multiplies low1
    cmap[2 * NMAP + base] = s3 - s2;    // multiplies low2
    cmap[3 * NMAP + base] = 1.0f - s3;  // multiplies low3
}

// ---------------------------------------------------------------- WMMA 16x16 tile GEMM (K=64)
// Arow: per-lane pointer to A[m= ti*16+lr][k= lh*2], row stride SX.
// Bcol: per-lane pointer to Bt[n= tj*16+lr][k= lh*2]  (B supplied k-major per column:
//       either a symmetric matrix read row-wise, or an explicitly transposed tile).
// All accesses: contiguous even-offset float2 -> ds_load_b64, banks = 4*lr mod 64 (no conflicts).
__device__ __forceinline__ v8f gemm_tile64(const float* __restrict__ Arow,
                                           const float* __restrict__ Bcol) {
    v8f c = {};
    #pragma unroll
    for (int kk = 0; kk < 16; ++kk) {
        v2f a = *(const v2f*)(Arow + kk * 4);
        v2f b = *(const v2f*)(Bcol + kk * 4);
        c = __builtin_amdgcn_wmma_f32_16x16x4_f32(false, a, false, b,
                                                  (short)0, c, false, false);
    }
    return c;
}

// ---------------------------------------------------------------- K2: per-image band filtering
// grid: 8192 blocks (one 64x64 image), 256 threads = 8 waves; each wave owns 2 tiles
__global__ void __launch_bounds__(256)
fdconv_main_kernel(const float* __restrict__ x,
                   const float* __restrict__ dD,
                   const float* __restrict__ cmap,
                   float* __restrict__ out) {
    __shared__ float xs[HW * SX];         // X, padded
    __shared__ float tsT[HW * SX];        // T transposed: tsT[n][k]
    __shared__ float dml[3][HW * SX];     // D1..D3, padded (symmetric)

    const int img  = blockIdx.x;          // b*256 + c
    const int b    = img >> 8;
    const int cch  = img & 255;
    const int g    = cch >> 5;
    const int tid  = threadIdx.x;
    const int lane = tid & 31;
    const int wave = tid >> 5;
    const int lr   = lane & 15;
    const int lh   = lane >> 4;

    const int t0  = wave * 2;
    const int ti0 = t0 >> 2, tj0 = t0 & 3;
    const int ti1 = (t0 + 1) >> 2, tj1 = (t0 + 1) & 3;

    // ---- stage X and all three D matrices into padded LDS
    {
        const v4f* __restrict__ xsrc = (const v4f*)(x + (size_t)img * PIX);
        #pragma unroll
        for (int i = 0; i < 4; ++i) {
            int e = tid + i * 256;            // v4f index; 16 per row, no row crossing
            v4f gv = xsrc[e];
            int f = e * 4;
            float* p = &xs[(f >> 6) * SX + (f & 63)];
            v2f lo; lo.x = gv.x; lo.y = gv.y;
            v2f hi; hi.x = gv.z; hi.y = gv.w;
            *(v2f*)p = lo; *(v2f*)(p + 2) = hi;
        }
        #pragma unroll
        for (int bd = 0; bd < 3; ++bd) {
            const v4f* __restrict__ dsrc = (const v4f*)(dD + bd * PIX);
            #pragma unroll
            for (int i = 0; i < 4; ++i) {
                int e = tid + i * 256;
                v4f gv = dsrc[e];
                int f = e * 4;
                float* p = &dml[bd][(f >> 6) * SX + (f & 63)];
                v2f lo; lo.x = gv.x; lo.y = gv.y;
                v2f hi; hi.x = gv.z; hi.y = gv.w;
                *(v2f*)p = lo; *(v2f*)(p + 2) = hi;
            }
        }
    }
    __syncthreads();

    const int mapbase = (b * GROUPS + g) * PIX;
    // owned-tile pixel coordinates (C/D WMMA layout): row = ti*16 + v + 8*lh, col = tj*16 + lr
    const int r0 = ti0 * 16 + 8 * lh, c0 = tj0 * 16 + lr;
    const int r1 = ti1 * 16 + 8 * lh, c1 = tj1 * 16 + lr;

    // acc = s1 * x  for the two owned tiles
    v8f acc0, acc1;
    {
        const float* cm0 = cmap + mapbase;   // plane 0
        #pragma unroll
        for (int v = 0; v < 8; ++v) {
            acc0[v] = cm0[(r0 + v) * HW + c0] * xs[(r0 + v) * SX + c0];
            acc1[v] = cm0[(r1 + v) * HW + c1] * xs[(r1 + v) * SX + c1];
        }
    }

    for (int band = 0; band < 3; ++band) {
        const float* dmb = dml[band];
        // T = X * D : A = xs rows, B via symmetry = D rows (k-contiguous)
        v8f tA = gemm_tile64(&xs[(ti0 * 16 + lr) * SX + lh * 2], &dmb[(tj0 * 16 + lr) * SX + lh * 2]);
        v8f tB = gemm_tile64(&xs[(ti1 * 16 + lr) * SX + lh * 2], &dmb[(tj1 * 16 + lr) * SX + lh * 2]);

        __syncthreads();           // previous band's tsT reads complete
        // store T transposed: lane writes 8 contiguous floats per tile (b64 pairs)
        {
            float* p0 = &tsT[(tj0 * 16 + lr) * SX + ti0 * 16 + 8 * lh];
            float* p1 = &tsT[(tj1 * 16 + lr) * SX + ti1 * 16 + 8 * lh];
            #pragma unroll
            for (int v = 0; v < 8; v += 2) {
                v2f a; a.x = tA[v]; a.y = tA[v + 1];
                v2f bb; bb.x = tB[v]; bb.y = tB[v + 1];
                *(v2f*)(p0 + v) = a;
                *(v2f*)(p1 + v) = bb;
            }
        }
        __syncthreads();

        // low = D * T : A = D rows, B = tsT rows (already k-major per column)
        v8f lA = gemm_tile64(&dmb[(ti0 * 16 + lr) * SX + lh * 2], &tsT[(tj0 * 16 + lr) * SX + lh * 2]);
        v8f lB = gemm_tile64(&dmb[(ti1 * 16 + lr) * SX + lh * 2], &tsT[(tj1 * 16 + lr) * SX + lh * 2]);

        const float* cb = cmap + (size_t)(band + 1) * NMAP + mapbase;
        #pragma unroll
        for (int v = 0; v < 8; ++v) {
            acc0[v] = fmaf(cb[(r0 + v) * HW + c0], lA[v], acc0[v]);
            acc1[v] = fmaf(cb[(r1 + v) * HW + c1], lB[v], acc1[v]);
        }
    }

    float* o = out + (size_t)img * PIX;
    #pragma unroll
    for (int v = 0; v < 8; ++v) {
        o[(r0 + v) * HW + c0] = acc0[v];
        o[(r1 + v) * HW + c1] = acc1[v];
    }
}

// ---------------------------------------------------------------- launcher
extern "C" void kernel_launch(void* const* d_in, const int* in_sizes, int n_in,
                              void* d_out, int out_size, void* d_ws, size_t ws_size,
                              hipStream_t stream) {
    const float* x    = (const float*)d_in[0];   // [32,256,64,64]
    const float* w    = (const float*)d_in[1];   // [3,8,32,3,3]
    const float* bias = (const float*)d_in[2];   // [3,8]
    float* out = (float*)d_out;

    float* dD   = (float*)d_ws;                  // 3*4096 floats
    float* cmap = dD + 3 * PIX;                  // 4 * 1M floats (16 MB)

    build_dmat_kernel<<<(3 * PIX + 255) / 256, 256, 0, stream>>>(dD);
    conv_coeff_kernel<<<BATCH * GROUPS * 16, 256, 0, stream>>>(x, w, bias, cmap);
    fdconv_main_kernel<<<BATCH * CHAN, 256, 0, stream>>>(x, dD, cmap, out);
}